// GCN_47880295415877
// MI455X (gfx1250) — compile-verified
//
#include <hip/hip_runtime.h>

typedef __attribute__((ext_vector_type(2))) float v2f;
typedef __attribute__((ext_vector_type(8))) float v8f;

#define N_NODES   100000
#define N_EDGES   3200000
#define N_GRAPHS  2000
#define DIM       64
#define N_LAYERS  3

// ---------------------------------------------------------------------------
// 1) degree accumulation (edges only; self-loop handled as +1 in dinv)
// ---------------------------------------------------------------------------
__global__ __launch_bounds__(256) void deg_kernel(const long long* __restrict__ dst,
                                                  float* __restrict__ deg) {
    int e = blockIdx.x * blockDim.x + threadIdx.x;
    if (e >= N_EDGES) return;
    atomicAdd(&deg[(int)dst[e]], 1.0f);
}

__global__ __launch_bounds__(256) void dinv_kernel(const float* __restrict__ deg,
                                                   float* __restrict__ dinv) {
    int i = blockIdx.x * blockDim.x + threadIdx.x;
    if (i >= N_NODES) return;
    dinv[i] = rsqrtf(deg[i] + 1.0f);   // +1 = self loop; always > 0
}

// ---------------------------------------------------------------------------
// 2) embedding: h = x @ emb_w^T + emb_b   (N x 4) * (4 x 64)
//    One wave -> 16 rows; 4 col tiles; one V_WMMA_F32_16X16X4_F32 each.
// ---------------------------------------------------------------------------
__global__ __launch_bounds__(256) void embed_kernel(const float* __restrict__ x,
                                                    const float* __restrict__ emb_w,
                                                    const float* __restrict__ emb_b,
                                                    float* __restrict__ h) {
    int wave = (blockIdx.x * blockDim.x + threadIdx.x) >> 5;
    int lane = threadIdx.x & 31;
    int m0   = wave * 16;
    if (m0 >= N_NODES) return;          // uniform per wave -> EXEC all-1s inside
    const bool hi = lane >= 16;
    const int  ln = lane & 15;
    const int  row = m0 + ln;

    // A: 16x4 fp32. lanes 0-15: K=0,1 ; lanes 16-31: K=2,3
    v2f a;
    a.x = x[row * 4 + (hi ? 2 : 0)];
    a.y = x[row * 4 + (hi ? 3 : 1)];

    for (int n0 = 0; n0 < DIM; n0 += 16) {
        int n = n0 + ln;
        // B: 4x16, B[k][n] = emb_w[n][k]
        v2f b;
        b.x = emb_w[n * 4 + (hi ? 2 : 0)];
        b.y = emb_w[n * 4 + (hi ? 3 : 1)];
        float bias = emb_b[n];
        v8f c = {bias, bias, bias, bias, bias, bias, bias, bias};
        c = __builtin_amdgcn_wmma_f32_16x16x4_f32(false, a, false, b,
                                                  (short)0, c, false, false);
#pragma unroll
        for (int r = 0; r < 8; ++r)
            h[(m0 + r + (hi ? 8 : 0)) * DIM + n] = c[r];
    }
}

// ---------------------------------------------------------------------------
// 3) conv GEMM: t = h @ W^T  (N x 64)*(64 x 64), W staged in LDS (16 KB).
//    One wave -> 16x64 strip (4 accumulators), 64 fp32 WMMAs, A reused 4x.
//    Epilogue fuses the self-loop term: acc = t * dinv^2 (no separate pass).
// ---------------------------------------------------------------------------
__global__ __launch_bounds__(256) void conv_gemm_kernel(const float* __restrict__ h,
                                                        const float* __restrict__ W,
                                                        const float* __restrict__ dinv,
                                                        float* __restrict__ t,
                                                        float* __restrict__ acc) {
    __shared__ float Wl[DIM * DIM];
    for (int i = threadIdx.x; i < DIM * DIM; i += blockDim.x) Wl[i] = W[i];
    __syncthreads();

    const int wave = blockIdx.x * 8 + (threadIdx.x >> 5);   // 6250 strips
    const int m0   = wave * 16;
    if (m0 >= N_NODES) return;
    const int  lane = threadIdx.x & 31;
    const bool hi   = lane >= 16;
    const int  ln   = lane & 15;
    const int  row  = m0 + ln;
    const int  koff = hi ? 2 : 0;

    v8f c0 = {}, c1 = {}, c2 = {}, c3 = {};
#pragma unroll
    for (int k = 0; k < DIM; k += 4) {
        v2f a  = *(const v2f*)&h[row * DIM + k + koff];
        v2f b0 = *(const v2f*)&Wl[(ln +  0) * DIM + k + koff];
        v2f b1 = *(const v2f*)&Wl[(ln + 16) * DIM + k + koff];
        v2f b2 = *(const v2f*)&Wl[(ln + 32) * DIM + k + koff];
        v2f b3 = *(const v2f*)&Wl[(ln + 48) * DIM + k + koff];
        c0 = __builtin_amdgcn_wmma_f32_16x16x4_f32(false, a, false, b0, (short)0, c0, false, false);
        c1 = __builtin_amdgcn_wmma_f32_16x16x4_f32(false, a, false, b1, (short)0, c1, false, false);
        c2 = __builtin_amdgcn_wmma_f32_16x16x4_f32(false, a, false, b2, (short)0, c2, false, false);
        c3 = __builtin_amdgcn_wmma_f32_16x16x4_f32(false, a, false, b3, (short)0, c3, false, false);
    }
#pragma unroll
    for (int r = 0; r < 8; ++r) {
        int   rr = m0 + r + (hi ? 8 : 0);
        float s  = dinv[rr];
        float ss = s * s;
        t  [rr * DIM + ln +  0] = c0[r];
        t  [rr * DIM + ln + 16] = c1[r];
        t  [rr * DIM + ln + 32] = c2[r];
        t  [rr * DIM + ln + 48] = c3[r];
        acc[rr * DIM + ln +  0] = c0[r] * ss;
        acc[rr * DIM + ln + 16] = c1[r] * ss;
        acc[rr * DIM + ln + 32] = c2[r] * ss;
        acc[rr * DIM + ln + 48] = c3[r] * ss;
    }
}

// ---------------------------------------------------------------------------
// 4) edge scatter: 32 threads/edge, float2 per lane, L2-resident f32 atomics
// ---------------------------------------------------------------------------
__global__ __launch_bounds__(256) void scatter_edges_kernel(const long long* __restrict__ src,
                                                            const long long* __restrict__ dst,
                                                            const float* __restrict__ dinv,
                                                            const float* __restrict__ t,
                                                            float* __restrict__ acc) {
    int tid = blockIdx.x * blockDim.x + threadIdx.x;   // E*32 = 102,400,000 < 2^31
    if (tid >= N_EDGES * 32) return;
    int e = tid >> 5;
    int d = (tid & 31) * 2;
    int s  = (int)src[e];
    int dd = (int)dst[e];
    float nrm = dinv[s] * dinv[dd];
    v2f tv = *(const v2f*)&t[s * DIM + d];
    atomicAdd(&acc[dd * DIM + d + 0], tv.x * nrm);
    atomicAdd(&acc[dd * DIM + d + 1], tv.y * nrm);
}

__global__ __launch_bounds__(256) void bias_relu_kernel(const float* __restrict__ acc,
                                                        const float* __restrict__ b,
                                                        float* __restrict__ h) {
    int i = blockIdx.x * blockDim.x + threadIdx.x;
    if (i >= N_NODES * DIM) return;
    float v = acc[i] + b[i & 63];
    h[i] = v > 0.0f ? v : 0.0f;
}

// ---------------------------------------------------------------------------
// 5) pooling: per-graph sums + counts via L2 atomics
// ---------------------------------------------------------------------------
__global__ __launch_bounds__(256) void pool_kernel(const float* __restrict__ h,
                                                   const long long* __restrict__ batch,
                                                   float* __restrict__ gsum,
                                                   float* __restrict__ cnt) {
    int tid = blockIdx.x * blockDim.x + threadIdx.x;
    if (tid >= N_NODES * DIM) return;
    int node = tid >> 6;
    int d    = tid & 63;
    int g    = (int)batch[node];
    atomicAdd(&gsum[g * DIM + d], h[tid]);
    if (d == 0) atomicAdd(&cnt[g], 1.0f);
}

// ---------------------------------------------------------------------------
// 6) MLP head: one thread per graph (2000 threads, negligible cost)
// ---------------------------------------------------------------------------
__global__ __launch_bounds__(256) void head_kernel(const float* __restrict__ gsum,
                                                   const float* __restrict__ cnt,
                                                   const float* __restrict__ w1,
                                                   const float* __restrict__ b1,
                                                   const float* __restrict__ w2,
                                                   const float* __restrict__ b2,
                                                   float* __restrict__ out) {
    int g = blockIdx.x * blockDim.x + threadIdx.x;
    if (g >= N_GRAPHS) return;
    float inv = 1.0f / fmaxf(cnt[g], 1.0f);
    float o0 = b2[0], o1 = b2[1], o2 = b2[2];
    for (int j = 0; j < DIM; ++j) {
        float s = 0.0f;
        const float* gr = &gsum[g * DIM];
        const float* wr = &w1[j * DIM];
        for (int d = 0; d < DIM; ++d) s += gr[d] * wr[d];
        float p = fmaxf(s * inv + b1[j], 0.0f);
        o0 += p * w2[0 * DIM + j];
        o1 += p * w2[1 * DIM + j];
        o2 += p * w2[2 * DIM + j];
    }
    out[g * 3 + 0] = o0;
    out[g * 3 + 1] = o1;
    out[g * 3 + 2] = o2;
}

// ---------------------------------------------------------------------------
extern "C" void kernel_launch(void* const* d_in, const int* in_sizes, int n_in,
                              void* d_out, int out_size, void* d_ws, size_t ws_size,
                              hipStream_t stream) {
    const float*     x      = (const float*)d_in[0];
    const long long* eidx   = (const long long*)d_in[1];  // int64 per reference
    const long long* batch  = (const long long*)d_in[2];
    const float*     emb_w  = (const float*)d_in[3];
    const float*     emb_b  = (const float*)d_in[4];
    const float*     conv_w = (const float*)d_in[5];      // [3,64,64]
    const float*     conv_b = (const float*)d_in[6];      // [3,64]
    const float*     lin1_w = (const float*)d_in[7];
    const float*     lin1_b = (const float*)d_in[8];
    const float*     lin2_w = (const float*)d_in[9];
    const float*     lin2_b = (const float*)d_in[10];
    float*           out    = (float*)d_out;

    const long long* src = eidx;
    const long long* dst = eidx + N_EDGES;

    // workspace carve-up (floats); total ~78 MB
    float* ws   = (float*)d_ws;
    size_t off  = 0;
    float* deg  = ws + off; off += N_NODES;
    float* dinv = ws + off; off += N_NODES;
    float* h    = ws + off; off += (size_t)N_NODES * DIM;
    float* t    = ws + off; off += (size_t)N_NODES * DIM;
    float* acc  = ws + off; off += (size_t)N_NODES * DIM;
    float* gsum = ws + off; off += (size_t)N_GRAPHS * DIM;
    float* cnt  = ws + off; off += N_GRAPHS;

    // --- normalization ---
    hipMemsetAsync(deg, 0, N_NODES * sizeof(float), stream);
    deg_kernel<<<(N_EDGES + 255) / 256, 256, 0, stream>>>(dst, deg);
    dinv_kernel<<<(N_NODES + 255) / 256, 256, 0, stream>>>(deg, dinv);

    // --- embedding (WMMA fp32 16x16x4) ---
    {
        int waves  = (N_NODES + 15) / 16;        // 6250
        int blocks = (waves * 32 + 255) / 256;   // 782
        embed_kernel<<<blocks, 256, 0, stream>>>(x, emb_w, emb_b, h);
    }

    // --- GCN layers ---
    const int GEMM_BLOCKS = ((N_NODES + 15) / 16 + 7) / 8;          // 782
    const int NE = (N_NODES * DIM + 255) / 256;                     // 25000
    const int EE = (N_EDGES * 32 + 255) / 256;                      // 400000
    for (int l = 0; l < N_LAYERS; ++l) {
        conv_gemm_kernel<<<GEMM_BLOCKS, 256, 0, stream>>>(
            h, conv_w + (size_t)l * DIM * DIM, dinv, t, acc);
        scatter_edges_kernel<<<EE, 256, 0, stream>>>(src, dst, dinv, t, acc);
        bias_relu_kernel<<<NE, 256, 0, stream>>>(acc, conv_b + (size_t)l * DIM, h);
    }

    // --- pooling (gsum and cnt are contiguous in ws) ---
    hipMemsetAsync(gsum, 0, (size_t)(N_GRAPHS * DIM + N_GRAPHS) * sizeof(float), stream);
    pool_kernel<<<NE, 256, 0, stream>>>(h, batch, gsum, cnt);

    // --- head ---
    head_kernel<<<(N_GRAPHS + 255) / 256, 256, 0, stream>>>(
        gsum, cnt, lin1_w, lin1_b, lin2_w, lin2_b, out);
}